// SPE_ATSA_58780922413323
// MI455X (gfx1250) — compile-verified
//
#include <hip/hip_runtime.h>

typedef __attribute__((ext_vector_type(2))) float v2f;
typedef __attribute__((ext_vector_type(8))) float v8f;

#define SCALE_F 0.35355339059327373f  // (64/8)^-0.5

__device__ __forceinline__ v8f wmma4(v2f a, v2f b, v8f c) {
  // D = A(16x4,f32) * B(4x16,f32) + C(16x16,f32)
  return __builtin_amdgcn_wmma_f32_16x16x4_f32(false, a, false, b, (short)0, c,
                                               false, false);
}

// ---------------------------------------------------------------------------
// Kernel A: fused roll + 1x1 qkv conv + 3x3 depthwise conv.
// One block per (window, 64-channel chunk, batch); 10x10 halo tile in LDS.
// ---------------------------------------------------------------------------
__global__ void __launch_bounds__(256) qkv_dw_kernel(
    const float* __restrict__ x, const float* __restrict__ w_qkv,
    const float* __restrict__ w_dw, float* __restrict__ qkv_out) {
  extern __shared__ float smem[];
  float* xs  = smem;             // [64][100]  input channels, halo pixels
  float* qs  = xs + 64 * 100;    // [64][100]  qkv chunk at halo pixels
  float* wqs = qs + 64 * 100;    // [64][65]   weight chunk (padded)

  const int t   = threadIdx.x;
  const int win = blockIdx.x;
  const int o0  = blockIdx.y * 64;
  const int b   = blockIdx.z;
  const int wy = win >> 5, wx = win & 31;

  // Load x halo (rolled coordinates; zero pad outside rolled image).
  for (int e = t; e < 64 * 100; e += 256) {
    int c = e / 100, pp = e % 100;
    int hy = pp / 10, hx = pp % 10;
    int yy = wy * 8 - 1 + hy, xx = wx * 8 - 1 + hx;
    float v = 0.f;
    if (yy >= 0 && yy < 256 && xx >= 0 && xx < 256)
      v = x[((b * 64 + c) * 256 + ((yy - 4) & 255)) * 256 + ((xx - 4) & 255)];
    xs[c * 100 + pp] = v;
  }
  for (int e = t; e < 64 * 64; e += 256) {
    int o = e >> 6, c = e & 63;
    wqs[o * 65 + c] = w_qkv[(o0 + o) * 64 + c];
  }
  __syncthreads();

  const int lane = t & 31;
  const int wv   = t >> 5;
  const int m    = lane & 15;
  const int koff = (lane < 16) ? 0 : 2;
  const int hi   = (lane < 16) ? 0 : 8;

  // GEMM: qs[64 x 0..95] = wqs(64x64) * xs(64x100), WMMA tiles 4x6.
  for (int j = 0; j < 3; ++j) {
    int tau = wv + 8 * j;  // 0..23
    int mt = tau / 6, nt = tau % 6;
    v8f acc = {};
#pragma unroll
    for (int kk = 0; kk < 16; ++kk) {
      v2f a, bb;
      a.x  = wqs[(mt * 16 + m) * 65 + kk * 4 + koff];
      a.y  = wqs[(mt * 16 + m) * 65 + kk * 4 + koff + 1];
      bb.x = xs[(kk * 4 + koff) * 100 + nt * 16 + m];
      bb.y = xs[(kk * 4 + koff + 1) * 100 + nt * 16 + m];
      acc = wmma4(a, bb, acc);
    }
#pragma unroll
    for (int r = 0; r < 8; ++r)
      qs[(mt * 16 + r + hi) * 100 + nt * 16 + m] = acc[r];
  }
  // Remainder halo columns 96..99 (VALU).
  {
    int o = t >> 2, p = 96 + (t & 3);
    float acc = 0.f;
    for (int c = 0; c < 64; ++c) acc += wqs[o * 65 + c] * xs[c * 100 + p];
    qs[o * 100 + p] = acc;
  }
  __syncthreads();

  // Depthwise 3x3 on the 8x8 interior, write qkv (rolled coords).
  {
    int o = t >> 2, g = t & 3;
    float wd[9];
#pragma unroll
    for (int k = 0; k < 9; ++k) wd[k] = w_dw[(o0 + o) * 9 + k];
#pragma unroll
    for (int e = 0; e < 16; ++e) {
      int d = g * 16 + e;
      int iy = d >> 3, ix = d & 7;
      float acc = 0.f;
#pragma unroll
      for (int ky = 0; ky < 3; ++ky)
#pragma unroll
        for (int kx = 0; kx < 3; ++kx)
          acc += wd[ky * 3 + kx] * qs[o * 100 + (iy + ky) * 10 + (ix + kx)];
      qkv_out[((b * 192 + o0 + o) * 256 + wy * 8 + iy) * 256 + wx * 8 + ix] = acc;
    }
  }
}

// ---------------------------------------------------------------------------
// Kernel B: per-window channel attention. One block per window.
// thr branch uses the algebraic identity: mean_p conv3x3(q) is a 9-shift
// masked-sum contraction (collapses 19 GFLOP of conv into a tiny matvec).
// ---------------------------------------------------------------------------
__global__ void __launch_bounds__(256) attn_kernel(
    const float* __restrict__ qkv, const float* __restrict__ w_thr,
    const float* __restrict__ b_thr, const float* __restrict__ pos_emb,
    const float* __restrict__ w_blend, float* __restrict__ out_img) {
  extern __shared__ float smem[];
  float* qs   = smem;             // [64][65] unscaled q
  float* ks   = qs + 64 * 65;
  float* vs   = ks + 64 * 65;
  float* pes  = vs + 64 * 65;
  float* sims = pes + 64 * 65;
  float* sths = sims + 64 * 65;
  float* Ssum = sths + 64 * 65;   // [64][9]
  float* thrv = Ssum + 64 * 9;    // [64]
  float* red  = thrv + 64;        // [256]

  const int t = threadIdx.x;
  const int win = blockIdx.x;
  const int b = blockIdx.z;
  const int wy = win >> 5, wx = win & 31;
  const int y0 = wy * 8, x0 = wx * 8;

  // Load q,k,v windows (channel-major, padded stride 65).
  {
    int c = t >> 2, seg = t & 3;
    for (int e = 0; e < 16; ++e) {
      int d = seg * 16 + e;
      int gi = ((b * 192 + c) * 256 + y0 + (d >> 3)) * 256 + x0 + (d & 7);
      qs[c * 65 + d] = qkv[gi];
      ks[c * 65 + d] = qkv[gi + 64 * 256 * 256];
      vs[c * 65 + d] = qkv[gi + 128 * 256 * 256];
    }
  }
  for (int e = t; e < 4096; e += 256) pes[(e >> 6) * 65 + (e & 63)] = pos_emb[e];
  __syncthreads();

  // Per-channel shift sums S(c, ky, kx) for the SAME-padded 3x3 conv mean.
  if (t < 64) {
    const float* q = qs + t * 65;
    float T = 0, r0 = 0, r7 = 0, c0 = 0, c7 = 0;
    for (int d = 0; d < 64; ++d) T += q[d];
    for (int i = 0; i < 8; ++i) {
      r0 += q[i]; r7 += q[56 + i]; c0 += q[i * 8]; c7 += q[i * 8 + 7];
    }
    float q00 = q[0], q07 = q[7], q70 = q[56], q77 = q[63];
#pragma unroll
    for (int ky = 0; ky < 3; ++ky)
#pragma unroll
      for (int kx = 0; kx < 3; ++kx) {
        int dy = ky - 1, dx = kx - 1;
        float er = (dy < 0) ? r7 : (dy > 0 ? r0 : 0.f);
        float ec = (dx < 0) ? c7 : (dx > 0 ? c0 : 0.f);
        float cr = 0.f;
        if (dy < 0 && dx < 0) cr = q77;
        else if (dy < 0 && dx > 0) cr = q70;
        else if (dy > 0 && dx < 0) cr = q07;
        else if (dy > 0 && dx > 0) cr = q00;
        Ssum[t * 9 + ky * 3 + kx] = T - er - ec + cr;
      }
  }
  __syncthreads();
  if (t < 64) {
    float acc = 0.f;
    for (int c = 0; c < 64; ++c)
#pragma unroll
      for (int k = 0; k < 9; ++k)
        acc += w_thr[(t * 64 + c) * 9 + k] * Ssum[c * 9 + k];
    float v = acc * (1.f / 64.f) + b_thr[t];
    thrv[t] = (v < 0.f) ? 0.01f * v : v;  // leaky_relu(0.01)
  }
  __syncthreads();

  const int lane = t & 31;
  const int wv   = t >> 5;
  const int m    = lane & 15;
  const int koff = (lane < 16) ? 0 : 2;
  const int hi   = (lane < 16) ? 0 : 8;

  // sim = scale*(q @ k^T); scale folded into epilogue.
#pragma unroll
  for (int j = 0; j < 2; ++j) {
    int tau = wv + 8 * j;  // 0..15
    int mt = tau >> 2, nt = tau & 3;
    v8f acc = {};
#pragma unroll
    for (int kk = 0; kk < 16; ++kk) {
      v2f a, bb;
      a.x  = qs[(mt * 16 + m) * 65 + kk * 4 + koff];
      a.y  = qs[(mt * 16 + m) * 65 + kk * 4 + koff + 1];
      bb.x = ks[(nt * 16 + m) * 65 + kk * 4 + koff];
      bb.y = ks[(nt * 16 + m) * 65 + kk * 4 + koff + 1];
      acc = wmma4(a, bb, acc);
    }
#pragma unroll
    for (int r = 0; r < 8; ++r) {
      int i = mt * 16 + r + hi;
      int jj = nt * 16 + m;
      float s = acc[r] * SCALE_F;
      float pe = pes[i * 65 + jj];
      sims[i * 65 + jj] = s + pe;
      sths[i * 65 + jj] = ((s < thrv[i]) ? 0.f : s) + pe;
    }
  }
  __syncthreads();

  // Row softmax over j, then blend into sims (the attn matrix).
  {
    int row = t >> 2, seg = t & 3;
    float mx = -3.4e38f;
    for (int e = 0; e < 16; ++e) mx = fmaxf(mx, sims[row * 65 + seg * 16 + e]);
    red[t] = mx;
    __syncthreads();
    float rm = fmaxf(fmaxf(red[row * 4], red[row * 4 + 1]),
                     fmaxf(red[row * 4 + 2], red[row * 4 + 3]));
    __syncthreads();
    float sm = 0.f;
    for (int e = 0; e < 16; ++e)
      sm += __expf(sims[row * 65 + seg * 16 + e] - rm);
    red[t] = sm;
    __syncthreads();
    float rs = red[row * 4] + red[row * 4 + 1] + red[row * 4 + 2] + red[row * 4 + 3];
    float wb = w_blend[0];
    float inv = 1.f / rs;
    for (int e = 0; e < 16; ++e) {
      int idx = row * 65 + seg * 16 + e;
      float sf = __expf(sims[idx] - rm) * inv;
      sims[idx] = sf * wb + sths[idx] * (1.f - wb);
    }
  }
  __syncthreads();

  // out = attn @ v, store to rolled-coords image.
#pragma unroll
  for (int j = 0; j < 2; ++j) {
    int tau = wv + 8 * j;
    int mt = tau >> 2, nt = tau & 3;
    v8f acc = {};
#pragma unroll
    for (int kk = 0; kk < 16; ++kk) {
      v2f a, bb;
      a.x  = sims[(mt * 16 + m) * 65 + kk * 4 + koff];
      a.y  = sims[(mt * 16 + m) * 65 + kk * 4 + koff + 1];
      bb.x = vs[(kk * 4 + koff) * 65 + nt * 16 + m];
      bb.y = vs[(kk * 4 + koff + 1) * 65 + nt * 16 + m];
      acc = wmma4(a, bb, acc);
    }
#pragma unroll
    for (int r = 0; r < 8; ++r) {
      int i = mt * 16 + r + hi;
      int d = nt * 16 + m;
      out_img[((b * 64 + i) * 256 + y0 + (d >> 3)) * 256 + x0 + (d & 7)] = acc[r];
    }
  }
}

// ---------------------------------------------------------------------------
// Kernel C: 1x1 proj + bias + roll(-4,-4). One block per (b, h, 128 w pixels).
// ---------------------------------------------------------------------------
__global__ void __launch_bounds__(256) proj_kernel(
    const float* __restrict__ img, const float* __restrict__ w_proj,
    const float* __restrict__ b_proj, float* __restrict__ out) {
  extern __shared__ float smem[];
  float* xs  = smem;             // [64][128]
  float* wps = xs + 64 * 128;    // [64][65]

  const int t  = threadIdx.x;
  const int b  = blockIdx.z;
  const int h  = blockIdx.y;
  const int w0 = blockIdx.x * 128;

  {
    int c = t >> 1, half = t & 1;
    int hp = (h + 4) & 255;
    for (int e = 0; e < 64; ++e) {
      int p = half * 64 + e;
      int wp_ = (w0 + p + 4) & 255;
      xs[c * 128 + p] = img[((b * 64 + c) * 256 + hp) * 256 + wp_];
    }
  }
  for (int e = t; e < 4096; e += 256) wps[(e >> 6) * 65 + (e & 63)] = w_proj[e];
  __syncthreads();

  const int lane = t & 31;
  const int wv   = t >> 5;
  const int m    = lane & 15;
  const int koff = (lane < 16) ? 0 : 2;
  const int hi   = (lane < 16) ? 0 : 8;

#pragma unroll
  for (int j = 0; j < 4; ++j) {
    int tau = wv + 8 * j;  // 0..31
    int mt = tau >> 3, nt = tau & 7;
    v8f acc = {};
#pragma unroll
    for (int kk = 0; kk < 16; ++kk) {
      v2f a, bb;
      a.x  = wps[(mt * 16 + m) * 65 + kk * 4 + koff];
      a.y  = wps[(mt * 16 + m) * 65 + kk * 4 + koff + 1];
      bb.x = xs[(kk * 4 + koff) * 128 + nt * 16 + m];
      bb.y = xs[(kk * 4 + koff + 1) * 128 + nt * 16 + m];
      acc = wmma4(a, bb, acc);
    }
#pragma unroll
    for (int r = 0; r < 8; ++r) {
      int o = mt * 16 + r + hi;
      int p = nt * 16 + m;
      out[((b * 64 + o) * 256 + h) * 256 + w0 + p] = acc[r] + b_proj[o];
    }
  }
}

// ---------------------------------------------------------------------------
extern "C" void kernel_launch(void* const* d_in, const int* in_sizes, int n_in,
                              void* d_out, int out_size, void* d_ws,
                              size_t ws_size, hipStream_t stream) {
  const float* x       = (const float*)d_in[0];
  const float* w_qkv   = (const float*)d_in[1];
  const float* w_dw    = (const float*)d_in[2];
  const float* w_thr   = (const float*)d_in[3];
  const float* b_thr   = (const float*)d_in[4];
  const float* pos_emb = (const float*)d_in[5];
  const float* w_blend = (const float*)d_in[6];
  const float* w_proj  = (const float*)d_in[7];
  const float* b_proj  = (const float*)d_in[8];
  float* out = (float*)d_out;

  float* qkv  = (float*)d_ws;                        // [4,192,256,256] f32
  float* oimg = qkv + (size_t)4 * 192 * 256 * 256;   // [4, 64,256,256] f32

  size_t shA = (size_t)(64 * 100 + 64 * 100 + 64 * 65) * sizeof(float);
  size_t shB = (size_t)(6 * 64 * 65 + 64 * 9 + 64 + 256) * sizeof(float);
  size_t shC = (size_t)(64 * 128 + 64 * 65) * sizeof(float);

  (void)hipFuncSetAttribute((const void*)qkv_dw_kernel,
                            hipFuncAttributeMaxDynamicSharedMemorySize, (int)shA);
  (void)hipFuncSetAttribute((const void*)attn_kernel,
                            hipFuncAttributeMaxDynamicSharedMemorySize, (int)shB);
  (void)hipFuncSetAttribute((const void*)proj_kernel,
                            hipFuncAttributeMaxDynamicSharedMemorySize, (int)shC);

  qkv_dw_kernel<<<dim3(1024, 3, 4), 256, shA, stream>>>(x, w_qkv, w_dw, qkv);
  attn_kernel<<<dim3(1024, 1, 4), 256, shB, stream>>>(qkv, w_thr, b_thr,
                                                      pos_emb, w_blend, oimg);
  proj_kernel<<<dim3(2, 256, 4), 256, shC, stream>>>(oimg, w_proj, b_proj, out);
}